// MambaBlock_8555574854446
// MI455X (gfx1250) — compile-verified
//
#include <hip/hip_runtime.h>
#include <hip/hip_bf16.h>
#include <math.h>

typedef float v2f __attribute__((ext_vector_type(2)));
typedef float v8f __attribute__((ext_vector_type(8)));

#define D_MODEL_C 1024
#define D_STATE_C 16
#define D_CONV_C  4
#define D_INNER_C 2048
#define BATCH_C   4
#define SEQLEN_C  2048
#define NROW_C    (BATCH_C * SEQLEN_C)   // 8192 rows (b,l)
#define NCH_C     16                      // scan chunks per sequence
#define CHLEN_C   (SEQLEN_C / NCH_C)      // 128

#define A_STRIDE 20    // padded dword stride for As rows (conflict-free b64 frags)
#define B_STRIDE 132   // padded dword stride for Bs rows

__device__ __forceinline__ float silu_f(float x) {
  return x / (1.0f + expf(-x));
}

// Async global->LDS copy of 16 bytes per lane (GLOBAL_LOAD_ASYNC_TO_LDS_B128,
// tracked by ASYNCcnt). lptr: generic pointer to __shared__; low 32 bits of
// the flat LDS aperture address are the LDS byte offset.
__device__ __forceinline__ void async_copy_b128(const float* gptr, const float* lptr) {
  unsigned lds_off = (unsigned)(size_t)lptr;
  asm volatile("global_load_async_to_lds_b128 %0, %1, off"
               :: "v"(lds_off), "v"(gptr) : "memory");
}

// ---------------------------------------------------------------------------
// f32 WMMA GEMM: C[M,N] = A[M,K] @ B[K,N], row-major.
// Requires M%128==0, N%128==0, K%16==0 (true for all uses here).
// Block = 256 threads (8 waves). Macro tile 128x128, BK = 16.
// Wave grid 4(M) x 2(N); each wave owns 32x64 = 2x4 tiles of 16x16x4 WMMAs.
// Double-buffered LDS, staged with async global->LDS B128 copies so the next
// K-stage's HBM/L2 traffic overlaps the current stage's 32 WMMAs.
//   As[buf]: [128 rows][16 k] stride 20  -> A frags are aligned ds_load_b64
//   Bs[buf]: [16 k][128 cols] stride 132 -> B frags are 2x conflict-free b32
// ---------------------------------------------------------------------------
__global__ __launch_bounds__(256) void gemm_f32_wmma(
    const float* __restrict__ A, const float* __restrict__ B,
    float* __restrict__ C, int M, int N, int K)
{
  __shared__ float As[2][128 * A_STRIDE];
  __shared__ float Bs[2][16 * B_STRIDE];

  const int tid  = threadIdx.x;
  const int lane = tid & 31;
  const int wid  = tid >> 5;      // 0..7
  const int wm   = wid >> 1;      // 0..3  -> 32-row strip
  const int wn   = wid & 1;       // 0..1  -> 64-col strip
  const int hi   = lane >> 4;     // half-wave select (K pair {0,1} vs {2,3})
  const int lm   = lane & 15;
  const int bm   = blockIdx.y * 128;
  const int bn   = blockIdx.x * 128;

  // fixed per-thread staging coordinates (2 x A chunks + 2 x B chunks = 4
  // async B128 issues per thread per stage; 16KB/stage per workgroup)
  const int a_row0 = tid >> 2;            // 0..63
  const int a_row1 = a_row0 + 64;         // 64..127
  const int a_c4   = (tid & 3) * 4;       // k sub-offset {0,4,8,12}
  const int b_k0   = tid >> 5;            // 0..7
  const int b_k1   = b_k0 + 8;            // 8..15
  const int b_n4   = (tid & 31) * 4;      // col sub-offset

  v8f acc[2][4];
  for (int i = 0; i < 2; i++)
    for (int j = 0; j < 4; j++)
      for (int r = 0; r < 8; r++) acc[i][j][r] = 0.0f;

  auto issue_stage = [&](int s, int buf) {
    const int kt = s * 16;
    async_copy_b128(&A[(size_t)(bm + a_row0) * K + kt + a_c4],
                    &As[buf][a_row0 * A_STRIDE + a_c4]);
    async_copy_b128(&A[(size_t)(bm + a_row1) * K + kt + a_c4],
                    &As[buf][a_row1 * A_STRIDE + a_c4]);
    async_copy_b128(&B[(size_t)(kt + b_k0) * N + bn + b_n4],
                    &Bs[buf][b_k0 * B_STRIDE + b_n4]);
    async_copy_b128(&B[(size_t)(kt + b_k1) * N + bn + b_n4],
                    &Bs[buf][b_k1 * B_STRIDE + b_n4]);
  };

  const int S = K >> 4;
  issue_stage(0, 0);

  for (int s = 0; s < S; s++) {
    const int cur = s & 1;
    __syncthreads();   // all waves done computing on buf[cur^1] -> free to refill
    if (s + 1 < S) {
      issue_stage(s + 1, cur ^ 1);
      // 4 outstanding = the stage s+1 issues; in-order completion => stage s landed
      asm volatile("s_wait_asynccnt 4" ::: "memory");
    } else {
      asm volatile("s_wait_asynccnt 0" ::: "memory");
    }
    __syncthreads();   // stage-s LDS contents visible to all waves

    const float* as = As[cur];
    const float* bs = Bs[cur];
    #pragma unroll
    for (int k4 = 0; k4 < 4; k4++) {
      const int kk = k4 * 4 + hi * 2;  // lanes 0-15: K {0,1}; lanes 16-31: K {2,3}
      v2f a0 = *(const v2f*)&as[(wm * 32 +  0 + lm) * A_STRIDE + kk];
      v2f a1 = *(const v2f*)&as[(wm * 32 + 16 + lm) * A_STRIDE + kk];
      v2f b0, b1, b2, b3;
      const int nb = wn * 64 + lm;
      b0.x = bs[(kk    ) * B_STRIDE + nb +  0];
      b0.y = bs[(kk + 1) * B_STRIDE + nb +  0];
      b1.x = bs[(kk    ) * B_STRIDE + nb + 16];
      b1.y = bs[(kk + 1) * B_STRIDE + nb + 16];
      b2.x = bs[(kk    ) * B_STRIDE + nb + 32];
      b2.y = bs[(kk + 1) * B_STRIDE + nb + 32];
      b3.x = bs[(kk    ) * B_STRIDE + nb + 48];
      b3.y = bs[(kk + 1) * B_STRIDE + nb + 48];
      acc[0][0] = __builtin_amdgcn_wmma_f32_16x16x4_f32(false, a0, false, b0, (short)0, acc[0][0], false, false);
      acc[0][1] = __builtin_amdgcn_wmma_f32_16x16x4_f32(false, a0, false, b1, (short)0, acc[0][1], false, false);
      acc[0][2] = __builtin_amdgcn_wmma_f32_16x16x4_f32(false, a0, false, b2, (short)0, acc[0][2], false, false);
      acc[0][3] = __builtin_amdgcn_wmma_f32_16x16x4_f32(false, a0, false, b3, (short)0, acc[0][3], false, false);
      acc[1][0] = __builtin_amdgcn_wmma_f32_16x16x4_f32(false, a1, false, b0, (short)0, acc[1][0], false, false);
      acc[1][1] = __builtin_amdgcn_wmma_f32_16x16x4_f32(false, a1, false, b1, (short)0, acc[1][1], false, false);
      acc[1][2] = __builtin_amdgcn_wmma_f32_16x16x4_f32(false, a1, false, b2, (short)0, acc[1][2], false, false);
      acc[1][3] = __builtin_amdgcn_wmma_f32_16x16x4_f32(false, a1, false, b3, (short)0, acc[1][3], false, false);
    }
  }

  // ---- store C: VGPR r holds M = r (lanes 0-15) / M = r+8 (lanes 16-31), N = lm
  for (int mi = 0; mi < 2; mi++) {
    const int row0 = bm + wm * 32 + mi * 16 + hi * 8;
    for (int ni = 0; ni < 4; ni++) {
      const int col = bn + wn * 64 + ni * 16 + lm;
      #pragma unroll
      for (int r = 0; r < 8; r++)
        C[(size_t)(row0 + r) * N + col] = acc[mi][ni][r];
    }
  }
}

// ---------------------------------------------------------------------------
// Causal depthwise conv1d (K=4) on xc = xz[:, :, 0:2048], then SiLU -> u
// ---------------------------------------------------------------------------
__global__ void conv_silu_kernel(const float* __restrict__ xz,
                                 const float* __restrict__ conv_w,
                                 float* __restrict__ u)
{
  const int idx = blockIdx.x * blockDim.x + threadIdx.x;  // over B*L*D_INNER
  if (idx >= NROW_C * D_INNER_C) return;
  const int c = idx % D_INNER_C;
  const int r = idx / D_INNER_C;           // b*L + l
  const int l = r % SEQLEN_C;
  float accv = 0.0f;
  #pragma unroll
  for (int j = 0; j < D_CONV_C; j++) {
    const int t = l - (D_CONV_C - 1) + j;
    if (t >= 0)
      accv += conv_w[c * D_CONV_C + j] *
              xz[(size_t)(r - l + t) * (2 * D_INNER_C) + c];
  }
  u[idx] = silu_f(accv);
}

// ---------------------------------------------------------------------------
// xp[r][j] = sum_c u[r][c] * W_x[c][j]   (j < 32). 8 rows per 256-thread block.
// ---------------------------------------------------------------------------
__global__ void proj_x_kernel(const float* __restrict__ u,
                              const float* __restrict__ W_x,
                              float* __restrict__ xp)
{
  const int j  = threadIdx.x & 31;
  const int rr = threadIdx.x >> 5;
  const int r  = blockIdx.x * 8 + rr;
  if (r >= NROW_C) return;
  float accv = 0.0f;
  const float* urow = &u[(size_t)r * D_INNER_C];
  for (int c = 0; c < D_INNER_C; c++)
    accv += urow[c] * W_x[c * (2 * D_STATE_C) + j];
  xp[(size_t)r * (2 * D_STATE_C) + j] = accv;
}

// ---------------------------------------------------------------------------
// dt = clip(softplus(dt_raw @ W_dt + b_dt)), Bp = B_raw @ W_Bp, ub = u * Bp
// ---------------------------------------------------------------------------
__global__ void dt_ub_kernel(const float* __restrict__ xp,
                             const float* __restrict__ W_dt,
                             const float* __restrict__ b_dt,
                             const float* __restrict__ W_Bp,
                             const float* __restrict__ u,
                             float* __restrict__ dt,
                             float* __restrict__ ub)
{
  const int idx = blockIdx.x * blockDim.x + threadIdx.x;
  if (idx >= NROW_C * D_INNER_C) return;
  const int c = idx % D_INNER_C;
  const int r = idx / D_INNER_C;
  const float* xpr = &xp[(size_t)r * (2 * D_STATE_C)];
  float dv = b_dt[c];
  float bp = 0.0f;
  #pragma unroll
  for (int s = 0; s < D_STATE_C; s++) {
    dv += xpr[s]             * W_dt[s * D_INNER_C + c];
    bp += xpr[D_STATE_C + s] * W_Bp[s * D_INNER_C + c];
  }
  float sp = (dv > 20.0f) ? dv : log1pf(expf(dv));
  sp = fminf(fmaxf(sp, 1e-4f), 1.0f);
  dt[idx] = sp;
  ub[idx] = u[idx] * bp;
}

// ---------------------------------------------------------------------------
// Scan pass 1: per (b, c, chunk), h0 = 0. In place:
//   ub[t] <- local h_t ; dt[t] <- prefix prod p_t of a = (1-dt).
// Also emit chunk summaries Ach (total decay), Xch (local final h).
// ---------------------------------------------------------------------------
__global__ void scan_pass1(float* __restrict__ dt, float* __restrict__ ub,
                           float* __restrict__ Ach, float* __restrict__ Xch)
{
  const int idx = blockIdx.x * blockDim.x + threadIdx.x;  // B*NCH*D_INNER
  if (idx >= BATCH_C * NCH_C * D_INNER_C) return;
  const int c  = idx % D_INNER_C;
  const int ch = (idx / D_INNER_C) % NCH_C;
  const int b  = idx / (D_INNER_C * NCH_C);
  size_t base = ((size_t)b * SEQLEN_C + (size_t)ch * CHLEN_C) * D_INNER_C + c;
  float h = 0.0f, p = 1.0f;
  for (int t = 0; t < CHLEN_C; t++) {
    const size_t o = base + (size_t)t * D_INNER_C;
    const float a = 1.0f - dt[o];
    h = a * h + ub[o];
    p *= a;
    ub[o] = h;   // local h
    dt[o] = p;   // prefix decay
  }
  const size_t so = ((size_t)b * NCH_C + ch) * D_INNER_C + c;
  Ach[so] = p;
  Xch[so] = h;
}

// ---------------------------------------------------------------------------
// Scan pass 2: per (b, c), serial over 16 chunks -> incoming state per chunk
// ---------------------------------------------------------------------------
__global__ void scan_pass2(const float* __restrict__ Ach,
                           const float* __restrict__ Xch,
                           float* __restrict__ Hin)
{
  const int idx = blockIdx.x * blockDim.x + threadIdx.x;  // B*D_INNER
  if (idx >= BATCH_C * D_INNER_C) return;
  const int c = idx % D_INNER_C;
  const int b = idx / D_INNER_C;
  float h = 0.0f;
  for (int ch = 0; ch < NCH_C; ch++) {
    const size_t so = ((size_t)b * NCH_C + ch) * D_INNER_C + c;
    Hin[so] = h;
    h = Ach[so] * h + Xch[so];
  }
}

// ---------------------------------------------------------------------------
// Scan pass 3 + epilogue: h_t = h_local + p_t * h_in; y = (h + D*u)*silu(z)
// Writes y over ub in place (becomes input to the output GEMM).
// ---------------------------------------------------------------------------
__global__ void scan_pass3(float* __restrict__ ub, const float* __restrict__ dt,
                           const float* __restrict__ Hin,
                           const float* __restrict__ u,
                           const float* __restrict__ xz,
                           const float* __restrict__ D_param)
{
  const int idx = blockIdx.x * blockDim.x + threadIdx.x;
  if (idx >= NROW_C * D_INNER_C) return;
  const int c = idx % D_INNER_C;
  const int r = idx / D_INNER_C;
  const int l = r % SEQLEN_C;
  const int b = r / SEQLEN_C;
  const int ch = l / CHLEN_C;
  const float hin = Hin[((size_t)b * NCH_C + ch) * D_INNER_C + c];
  const float h = ub[idx] + dt[idx] * hin;
  const float yv = h + D_param[c] * u[idx];
  const float zv = xz[(size_t)r * (2 * D_INNER_C) + D_INNER_C + c];
  ub[idx] = yv * silu_f(zv);
}

// ---------------------------------------------------------------------------
extern "C" void kernel_launch(void* const* d_in, const int* in_sizes, int n_in,
                              void* d_out, int out_size, void* d_ws, size_t ws_size,
                              hipStream_t stream) {
  const float* x       = (const float*)d_in[0];
  const float* W_in    = (const float*)d_in[1];
  const float* conv_w  = (const float*)d_in[2];
  const float* W_x     = (const float*)d_in[3];
  const float* W_dt    = (const float*)d_in[4];
  const float* b_dt    = (const float*)d_in[5];
  const float* W_Bp    = (const float*)d_in[6];
  const float* D_param = (const float*)d_in[7];
  const float* W_out   = (const float*)d_in[8];
  float* out = (float*)d_out;

  // workspace layout (floats)
  float* ws = (float*)d_ws;
  size_t off = 0;
  float* xz  = ws + off; off += (size_t)NROW_C * 2 * D_INNER_C;   // 33.5M
  float* u   = ws + off; off += (size_t)NROW_C * D_INNER_C;       // 16.8M
  float* xp  = ws + off; off += (size_t)NROW_C * 2 * D_STATE_C;   // 0.26M
  float* dt  = ws + off; off += (size_t)NROW_C * D_INNER_C;       // 16.8M
  float* ub  = ws + off; off += (size_t)NROW_C * D_INNER_C;       // 16.8M (-> y)
  float* Ach = ws + off; off += (size_t)BATCH_C * NCH_C * D_INNER_C;
  float* Xch = ws + off; off += (size_t)BATCH_C * NCH_C * D_INNER_C;
  float* Hin = ws + off; off += (size_t)BATCH_C * NCH_C * D_INNER_C;

  // 1) xz = x @ W_in          [8192,1024] x [1024,4096]
  gemm_f32_wmma<<<dim3(2 * D_INNER_C / 128, NROW_C / 128), 256, 0, stream>>>(
      x, W_in, xz, NROW_C, 2 * D_INNER_C, D_MODEL_C);

  // 2) u = silu(conv(xc))
  {
    const int n = NROW_C * D_INNER_C;
    conv_silu_kernel<<<(n + 255) / 256, 256, 0, stream>>>(xz, conv_w, u);
  }

  // 3) xp = u @ W_x           [8192,2048] x [2048,32]
  proj_x_kernel<<<NROW_C / 8, 256, 0, stream>>>(u, W_x, xp);

  // 4) dt, ub
  {
    const int n = NROW_C * D_INNER_C;
    dt_ub_kernel<<<(n + 255) / 256, 256, 0, stream>>>(xp, W_dt, b_dt, W_Bp, u, dt, ub);
  }

  // 5) chunked linear-recurrence scan (3 passes)
  {
    const int n1 = BATCH_C * NCH_C * D_INNER_C;
    scan_pass1<<<(n1 + 255) / 256, 256, 0, stream>>>(dt, ub, Ach, Xch);
    const int n2 = BATCH_C * D_INNER_C;
    scan_pass2<<<(n2 + 255) / 256, 256, 0, stream>>>(Ach, Xch, Hin);
    const int n3 = NROW_C * D_INNER_C;
    scan_pass3<<<(n3 + 255) / 256, 256, 0, stream>>>(ub, dt, Hin, u, xz, D_param);
  }

  // 6) out = y @ W_out        [8192,2048] x [2048,1024]  (y lives in ub)
  gemm_f32_wmma<<<dim3(D_MODEL_C / 128, NROW_C / 128), 256, 0, stream>>>(
      ub, W_out, out, NROW_C, D_MODEL_C, D_INNER_C);
}